// HungarianSumOfDistancesLoss_37666863186635
// MI455X (gfx1250) — compile-verified
//
#include <hip/hip_runtime.h>
#include <math.h>

typedef __attribute__((ext_vector_type(2))) float v2f;
typedef __attribute__((ext_vector_type(8))) float v8f;

#define NPTS  256
#define BATCH 64
#define DIM   3
#define FINF  1e30f

// ---------------------------------------------------------------------------
// Kernel 0: zero the scalar output (atomicAdd accumulator).
// ---------------------------------------------------------------------------
__global__ void zero_out_kernel(float* out) {
    if (blockIdx.x == 0 && threadIdx.x == 0) out[0] = 0.0f;
}

// ---------------------------------------------------------------------------
// Kernel 1: distance matrix via V_WMMA_F32_16X16X4_F32.
// One wave computes one 16x16 tile of dist[b] using a single WMMA:
//   A[m][k]  = { -2*p[k]  (k<3),  1      (k==3) }   (16x4)
//   B[k][n]  = {  t[k]    (k<3),  |t_n|^2 (k==3) }  (4x16)
//   C[m][n]  =   |p_m|^2  (broadcast per row via lane shuffle)
//   D = A*B + C = |p|^2 + |t|^2 - 2 p.t  ->  sqrt -> dist tile
// A layout (32b, 16x4): lanes 0-15 hold K={0,1}, lanes 16-31 K={2,3}, M=lane%16.
// C/D layout: VGPR r -> M = r + 8*(lane>=16), N = lane%16.
// ---------------------------------------------------------------------------
__global__ __launch_bounds__(32)
void dist_wmma_kernel(const float* __restrict__ pred,
                      const float* __restrict__ target,
                      float* __restrict__ dist) {
    const int lane = threadIdx.x & 31;
    const int half = lane >> 4;      // 0: K={0,1}  1: K={2,3}
    const int m    = lane & 15;      // A-row / B-col within tile
    const int tn   = blockIdx.x;     // tile column (N)
    const int tm   = blockIdx.y;     // tile row    (M)
    const int b    = blockIdx.z;

    const float* pr = pred   + ((size_t)b * NPTS + tm * 16 + m) * DIM;
    const float* tg = target + ((size_t)b * NPTS + tn * 16 + m) * DIM;

    const float p0 = pr[0], p1 = pr[1], p2 = pr[2];
    const float t0 = tg[0], t1 = tg[1], t2 = tg[2];
    const float normP = p0 * p0 + p1 * p1 + p2 * p2;
    const float normT = t0 * t0 + t1 * t1 + t2 * t2;

    v2f a, bm;
    if (half == 0) {
        a.x = -2.0f * p0;  a.y = -2.0f * p1;
        bm.x = t0;         bm.y = t1;
    } else {
        a.x = -2.0f * p2;  a.y = 1.0f;
        bm.x = t2;         bm.y = normT;
    }

    // C accumulator: VGPR r carries |p_M|^2 with M = r + 8*half; the lane that
    // loaded pred row M is lane==M (lanes 0..15 own rows 0..15).
    v8f c;
#pragma unroll
    for (int r = 0; r < 8; ++r)
        c[r] = __shfl(normP, r + 8 * half, 32);

    v8f d = __builtin_amdgcn_wmma_f32_16x16x4_f32(
        /*neg_a=*/false, a, /*neg_b=*/false, bm,
        /*c_mod=*/(short)0, c, /*reuse_a=*/false, /*reuse_b=*/false);

    float* out = dist + (size_t)b * NPTS * NPTS;
#pragma unroll
    for (int r = 0; r < 8; ++r) {
        const int M = tm * 16 + r + 8 * half;
        const int Ncol = tn * 16 + m;
        float v = d[r];
        out[M * NPTS + Ncol] = (v > 0.0f) ? sqrtf(v) : 0.0f;
    }
}

// ---------------------------------------------------------------------------
// Kernel 2: exact Hungarian (JV with potentials), one workgroup per batch,
// one thread per column. Parallel slack update + LDS argmin per path step;
// augmenting flip is serial (short) on thread 0. Matches the reference
// e-maxx algorithm structure exactly.
// ---------------------------------------------------------------------------
__global__ __launch_bounds__(NPTS)
void hungarian_kernel(const float* __restrict__ dist, float* __restrict__ out) {
    const int b   = blockIdx.x;
    const int tid = threadIdx.x;     // 0..255
    const int j   = tid + 1;         // column 1..256 (1-based)
    const float* D = dist + (size_t)b * NPTS * NPTS;

    __shared__ float u[NPTS + 1], v[NPTS + 1], minv[NPTS + 1];
    __shared__ int   p[NPTS + 1], way[NPTS + 1];
    __shared__ unsigned char used[NPTS + 1];
    __shared__ float red_val[NPTS];
    __shared__ int   red_idx[NPTS];
    __shared__ int   s_j0, s_done;

    u[j] = 0.0f; v[j] = 0.0f; p[j] = 0; way[j] = 0;
    if (tid == 0) { u[0] = 0.0f; v[0] = 0.0f; p[0] = 0; way[0] = 0; }
    __syncthreads();

    for (int i = 1; i <= NPTS; ++i) {
        if (tid == 0) { p[0] = i; s_j0 = 0; s_done = 0; used[0] = 0; }
        minv[j] = FINF;
        used[j] = 0;
        __syncthreads();

        while (true) {
            const int j0 = s_j0;
            if (tid == 0) used[j0] = 1;
            __syncthreads();

            const int   i0  = p[j0];
            const float ui0 = u[i0];
            if (!used[j]) {
                const float cur = D[(i0 - 1) * NPTS + (j - 1)] - ui0 - v[j];
                if (cur < minv[j]) { minv[j] = cur; way[j] = j0; }
            }
            __syncthreads();

            // argmin over unused columns
            red_val[tid] = used[j] ? FINF : minv[j];
            red_idx[tid] = j;
            __syncthreads();
#pragma unroll
            for (int s = NPTS / 2; s > 0; s >>= 1) {
                if (tid < s) {
                    if (red_val[tid + s] < red_val[tid]) {
                        red_val[tid] = red_val[tid + s];
                        red_idx[tid] = red_idx[tid + s];
                    }
                }
                __syncthreads();
            }
            const float delta = red_val[0];
            const int   j1    = red_idx[0];

            // potential update: used columns have distinct matched rows,
            // so the u[p[j]] writes never collide.
            if (used[j]) { u[p[j]] += delta; v[j] -= delta; }
            else         { minv[j] -= delta; }
            if (tid == 0) {
                if (used[0]) { u[p[0]] += delta; v[0] -= delta; }
                s_j0 = j1;
                if (p[j1] == 0) s_done = 1;
            }
            __syncthreads();
            if (s_done) break;
        }

        // augment along the alternating path (short; serial)
        if (tid == 0) {
            int jj = s_j0;
            while (jj) { const int jn = way[jj]; p[jj] = p[jn]; jj = jn; }
        }
        __syncthreads();
    }

    // sum matched costs: column j matched to row p[j]
    red_val[tid] = D[(p[j] - 1) * NPTS + (j - 1)];
    __syncthreads();
#pragma unroll
    for (int s = NPTS / 2; s > 0; s >>= 1) {
        if (tid < s) red_val[tid] += red_val[tid + s];
        __syncthreads();
    }
    if (tid == 0) atomicAdd(out, red_val[0] / (float)BATCH);
}

// ---------------------------------------------------------------------------
extern "C" void kernel_launch(void* const* d_in, const int* in_sizes, int n_in,
                              void* d_out, int out_size, void* d_ws, size_t ws_size,
                              hipStream_t stream) {
    const float* pred   = (const float*)d_in[0];   // [64, 256, 3] f32
    const float* target = (const float*)d_in[1];   // [64, 256, 3] f32
    float* out  = (float*)d_out;                   // scalar f32
    float* dist = (float*)d_ws;                    // 64*256*256 f32 = 16 MiB

    zero_out_kernel<<<1, 32, 0, stream>>>(out);

    dim3 g(NPTS / 16, NPTS / 16, BATCH);           // (16, 16, 64) tiles
    dist_wmma_kernel<<<g, 32, 0, stream>>>(pred, target, dist);

    hungarian_kernel<<<BATCH, NPTS, 0, stream>>>(dist, out);
}